// GConv_39376260170204
// MI455X (gfx1250) — compile-verified
//
#include <hip/hip_runtime.h>
#include <hip/hip_bf16.h>

// GCN 2-layer forward for MI455X (gfx1250, wave32).
// GEMM via V_WMMA_F32_16X16X4_F32 (exact fp32). Edge scatter via L2 float atomics
// (agg = 102MB fits in 192MB L2). D = H = 256 hardcoded.

typedef __attribute__((ext_vector_type(2))) float v2f;
typedef __attribute__((ext_vector_type(8))) float v8f;

#define DIMK 256
#define DIMH 256

__device__ __forceinline__ void atomAddF(float* p, float v) {
#if defined(__gfx1250__)
    unsafeAtomicAdd(p, v);   // hardware global_atomic_add_f32
#else
    atomicAdd(p, v);
#endif
}

// ---------------- degree / norm ----------------
__global__ void __launch_bounds__(256) init_deg_kernel(float* dinv, int n) {
    int i = blockIdx.x * 256 + threadIdx.x;
    if (i < n) dinv[i] = 1.0f;               // self-loop contributes 1
}

__global__ void __launch_bounds__(256) count_deg_kernel(const int* __restrict__ dst,
                                                        float* dinv, int e) {
    int i = blockIdx.x * 256 + threadIdx.x;
    if (i < e) atomAddF(&dinv[dst[i]], 1.0f);
}

__global__ void __launch_bounds__(256) rsqrt_deg_kernel(float* dinv, int n) {
    int i = blockIdx.x * 256 + threadIdx.x;
    if (i < n) dinv[i] = __frsqrt_rn(dinv[i]);   // deg >= 1 always
}

// ---------------- WMMA GEMM: Hout[N,256] = A[N,256] @ W[256,256] ----------------
// Block: 256 threads = 8 waves; tile 128 rows x 64 cols.
// LDS tile interleaved by K-pairs so each B fragment is ONE ds_load_b64 landing
// directly in an even-aligned VGPR pair (no v_dual_mov shuffles before WMMA):
//   Wt[(k>>1)*64 + c] = { W[k][colBase+c], W[k+1][colBase+c] }
__global__ void __launch_bounds__(256) gemm_wmma_kernel(const float* __restrict__ A,
                                                        const float* __restrict__ W,
                                                        float* __restrict__ Hout,
                                                        int nrows) {
    __shared__ v2f Wt[(DIMK / 2) * 64];      // 64 KB
    const int colBase = blockIdx.y * 64;
    for (int i = threadIdx.x; i < DIMK * 64; i += 256) {
        int k = i >> 6, c = i & 63;
        ((float*)Wt)[(((k >> 1) * 64 + c) << 1) + (k & 1)] = W[k * DIMH + colBase + c];
    }
    __syncthreads();

    const int wave = threadIdx.x >> 5;
    const int lane = threadIdx.x & 31;
    const int rowBase = (blockIdx.x * 8 + wave) * 16;
    if (rowBase >= nrows) return;            // wave-uniform: EXEC stays all-ones

    const int half = lane >> 4;              // 0: K pair (k,k+1)  1: (k+2,k+3)
    const int l16  = lane & 15;

    v8f acc0 = {}, acc1 = {}, acc2 = {}, acc3 = {};
    const float* arow = A + (size_t)(rowBase + l16) * DIMK;

    for (int k = 0; k < DIMK; k += 4) {
        const int ka = k + 2 * half;         // this lane's K pair base
        const int g  = ka >> 1;              // K-pair group index
        // A fragment (16x4 f32): lane l16 holds row rowBase+l16, K = ka, ka+1
        v2f a = *(const v2f*)(arow + ka);
        // B fragments (4x16 f32): single 8-byte LDS loads per column tile
        v2f b0 = Wt[g * 64 + l16 +  0];
        v2f b1 = Wt[g * 64 + l16 + 16];
        v2f b2 = Wt[g * 64 + l16 + 32];
        v2f b3 = Wt[g * 64 + l16 + 48];
        acc0 = __builtin_amdgcn_wmma_f32_16x16x4_f32(false, a, false, b0, (short)0, acc0, false, false);
        acc1 = __builtin_amdgcn_wmma_f32_16x16x4_f32(false, a, false, b1, (short)0, acc1, false, false);
        acc2 = __builtin_amdgcn_wmma_f32_16x16x4_f32(false, a, false, b2, (short)0, acc2, false, false);
        acc3 = __builtin_amdgcn_wmma_f32_16x16x4_f32(false, a, false, b3, (short)0, acc3, false, false);
    }

    // C/D layout: lanes 0-15 rows M=0..7 in v[0..7]; lanes 16-31 rows M=8..15
    float* o0 = Hout + (size_t)(rowBase + half * 8) * DIMH + colBase + l16;
    #pragma unroll
    for (int r = 0; r < 8; ++r) {
        o0[(size_t)r * DIMH +  0] = acc0[r];
        o0[(size_t)r * DIMH + 16] = acc1[r];
        o0[(size_t)r * DIMH + 32] = acc2[r];
        o0[(size_t)r * DIMH + 48] = acc3[r];
    }
}

// ---------------- agg init with self-loop: agg[v] = dinv[v]^2 * h[v] ----------------
__global__ void __launch_bounds__(256) init_agg_kernel(const float* __restrict__ h,
                                                       const float* __restrict__ dinv,
                                                       float* __restrict__ agg, int n) {
    size_t i = (size_t)blockIdx.x * 256 + threadIdx.x;     // over N*64 float4s
    size_t total = (size_t)n * (DIMH / 4);
    if (i >= total) return;
    int v = (int)(i >> 6);
    float w = dinv[v]; w *= w;
    float4 hv = ((const float4*)h)[i];
    float4 r; r.x = w * hv.x; r.y = w * hv.y; r.z = w * hv.z; r.w = w * hv.w;
    ((float4*)agg)[i] = r;
}

// ---------------- edge scatter: one wave per edge ----------------
__global__ void __launch_bounds__(256) scatter_kernel(const int* __restrict__ src,
                                                      const int* __restrict__ dst,
                                                      const float* __restrict__ dinv,
                                                      const float* __restrict__ h,
                                                      float* __restrict__ agg, int e) {
    int edge = blockIdx.x * 8 + (threadIdx.x >> 5);
    if (edge >= e) return;
    const int lane = threadIdx.x & 31;
    const int s = src[edge], d = dst[edge];
    const float w = dinv[s] * dinv[d];
    const float4* hs = (const float4*)(h + (size_t)s * DIMH);
    float* ad = agg + (size_t)d * DIMH;
    #pragma unroll
    for (int j = 0; j < 2; ++j) {
        const int q = lane + j * 32;         // float4 index within the 256-float row
        float4 v = hs[q];
        atomAddF(ad + q * 4 + 0, w * v.x);
        atomAddF(ad + q * 4 + 1, w * v.y);
        atomAddF(ad + q * 4 + 2, w * v.z);
        atomAddF(ad + q * 4 + 3, w * v.w);
    }
}

// ---------------- bias + ReLU (in place) ----------------
__global__ void __launch_bounds__(256) bias_relu_kernel(float* __restrict__ z,
                                                        const float* __restrict__ b, int n) {
    size_t i = (size_t)blockIdx.x * 256 + threadIdx.x;     // over N*64 float4s
    size_t total = (size_t)n * (DIMH / 4);
    if (i >= total) return;
    int c4 = (int)(i & 63);
    float4 bb = ((const float4*)b)[c4];
    float4 v = ((float4*)z)[i];
    v.x = fmaxf(v.x + bb.x, 0.0f);
    v.y = fmaxf(v.y + bb.y, 0.0f);
    v.z = fmaxf(v.z + bb.z, 0.0f);
    v.w = fmaxf(v.w + bb.w, 0.0f);
    ((float4*)z)[i] = v;
}

extern "C" void kernel_launch(void* const* d_in, const int* in_sizes, int n_in,
                              void* d_out, int out_size, void* d_ws, size_t ws_size,
                              hipStream_t stream) {
    const int E = in_sizes[0] / 2;
    const int N = in_sizes[1] / DIMK;

    const int*   ei  = (const int*)d_in[0];
    const int*   src = ei;
    const int*   dst = ei + E;
    const float* x   = (const float*)d_in[1];
    const float* W1  = (const float*)d_in[2];
    const float* b1  = (const float*)d_in[3];
    const float* W2  = (const float*)d_in[4];
    const float* b2  = (const float*)d_in[5];
    float* out = (float*)d_out;

    float* dinv = (float*)d_ws;              // N floats
    float* h    = dinv + N;                  // N*256 floats
    float* agg  = h + (size_t)N * DIMH;      // N*256 floats

    const int gN   = (N + 255) / 256;
    const int gE   = (E + 255) / 256;
    const int gNH4 = (int)(((size_t)N * (DIMH / 4) + 255) / 256);
    const int gEw  = (E + 7) / 8;            // one wave per edge
    dim3 gGemm((N + 127) / 128, DIMH / 64);

    // degree -> dinv
    init_deg_kernel <<<gN, 256, 0, stream>>>(dinv, N);
    count_deg_kernel<<<gE, 256, 0, stream>>>(dst, dinv, E);
    rsqrt_deg_kernel<<<gN, 256, 0, stream>>>(dinv, N);

    // layer 1: h = x@W1 ; agg = scatter(norm*h) ; agg = relu(agg+b1)
    gemm_wmma_kernel<<<gGemm, 256, 0, stream>>>(x, W1, h, N);
    init_agg_kernel <<<gNH4, 256, 0, stream>>>(h, dinv, agg, N);
    scatter_kernel  <<<gEw, 256, 0, stream>>>(src, dst, dinv, h, agg, E);
    bias_relu_kernel<<<gNH4, 256, 0, stream>>>(agg, b1, N);

    // layer 2: h = agg@W2 ; out = scatter(norm*h) ; out = relu(out+b2)
    gemm_wmma_kernel<<<gGemm, 256, 0, stream>>>(agg, W2, h, N);
    init_agg_kernel <<<gNH4, 256, 0, stream>>>(h, dinv, out, N);
    scatter_kernel  <<<gEw, 256, 0, stream>>>(src, dst, dinv, h, out, E);
    bias_relu_kernel<<<gNH4, 256, 0, stream>>>(out, b2, N);
}